// Transmodelv3_27934467293422
// MI455X (gfx1250) — compile-verified
//
#include <hip/hip_runtime.h>
#include <hip/hip_bf16.h>

// ---------------------------------------------------------------------------
// NeZha attention + MMoE for MI455X (gfx1250), fp32 via V_WMMA_F32_16X16X4_F32
// ---------------------------------------------------------------------------

typedef __attribute__((ext_vector_type(2))) float v2f;
typedef __attribute__((ext_vector_type(8))) float v8f;

#define ZERO8 {0.f,0.f,0.f,0.f,0.f,0.f,0.f,0.f}

__device__ __forceinline__ v8f wmma_f32(v2f a, v2f b, v8f c) {
  // 8 args: (neg_a, A, neg_b, B, c_mod, C, reuse_a, reuse_b)
  return __builtin_amdgcn_wmma_f32_16x16x4_f32(false, a, false, b, (short)0, c,
                                               false, false);
}

// Problem constants
#define BATCH 8
#define SEQ 512
#define HID 768
#define NH 12
#define HD 64
#define BH (BATCH * NH)       // 96
#define RVOC 129              // 2*MAX_REL+1
#define RPAD 144              // padded to multiple of 16
#define MMOE_DIM 128
#define N_TASK 4
#define N_EXPERT 3
#define DNN_DIM 128

// Workspace layout (floats)
#define OFF_TAB  ((size_t)0)                               //  144*64
#define OFF_Q    ((size_t)16384)
#define OFF_K    (OFF_Q + (size_t)BH * SEQ * HD)           // +3145728
#define OFF_V    (OFF_K + (size_t)BH * SEQ * HD)
#define OFF_QREL (OFF_V + (size_t)BH * SEQ * HD)
#define OFF_CTX  (OFF_QREL + (size_t)BH * SEQ * RPAD)      // +7077888
#define OFF_X    (OFF_CTX + (size_t)BATCH * SEQ * HID)

// ---------------------------------------------------------------------------
// Kernel 0: relative-position sinusoid table, [RPAD=144 rows, HD=64], rows
// >= 129 zero-filled so the padded K=144 GEMM contributes nothing.
// ---------------------------------------------------------------------------
__global__ void k_tab(float* __restrict__ tab) {
  int idx = blockIdx.x * 256 + threadIdx.x;
  if (idx >= RPAD * HD) return;
  int p = idx >> 6, d = idx & 63;
  float val = 0.f;
  if (p < RVOC) {
    int i = d >> 1;                            // pair index
    // div = exp(2i * (-ln(10000)/64))
    float div = __expf(-(float)(2 * i) * 0.14391156831212787f);
    float ang = (float)p * div;
    val = (d & 1) ? __cosf(ang) : __sinf(ang);
  }
  tab[idx] = val;
}

// ---------------------------------------------------------------------------
// Kernel 1: QKV projection.  C[4096,768] = hs[4096,768] @ W[768,768] + bias,
// scattered into head-major [BH, SEQ, HD].  One 16x16 tile per wave, K-loop
// in steps of 4 through V_WMMA_F32_16X16X4_F32.
// grid (256, 12, 3), block 128 (4 waves)
// ---------------------------------------------------------------------------
__global__ void k_qkv(const float* __restrict__ hs,
                      const float* __restrict__ Wq, const float* __restrict__ bq,
                      const float* __restrict__ Wk, const float* __restrict__ bk,
                      const float* __restrict__ Wv, const float* __restrict__ bv,
                      float* __restrict__ q, float* __restrict__ k,
                      float* __restrict__ v) {
  const int wave = threadIdx.x >> 5;
  const int lane = threadIdx.x & 31;
  const int lm = lane & 15, kh = lane >> 4;

  const int mt = blockIdx.x;                 // 0..255
  const int nt = blockIdx.y * 4 + wave;      // 0..47
  const int mat = blockIdx.z;                // 0=Q 1=K 2=V
  const float* W    = (mat == 0) ? Wq : (mat == 1) ? Wk : Wv;
  const float* bias = (mat == 0) ? bq : (mat == 1) ? bk : bv;
  float* dst        = (mat == 0) ? q  : (mat == 1) ? k  : v;

  const int m0 = mt * 16, n0 = nt * 16;
  const float* arow = hs + (size_t)(m0 + lm) * HID + 2 * kh;
  const int col = n0 + lm;

  v8f c = ZERO8;
  for (int kk = 0; kk < HID; kk += 4) {
    v2f a = *(const v2f*)(arow + kk);
    int kb = kk + 2 * kh;
    v2f b;
    b.x = W[(size_t)kb * HID + col];
    b.y = W[(size_t)(kb + 1) * HID + col];
    c = wmma_f32(a, b, c);
  }

  const float bval = bias[col];
  const int h = col >> 6;                    // head (16-col tile never straddles)
  const int d = col & 63;
#pragma unroll
  for (int r = 0; r < 8; ++r) {
    int row = m0 + r + 8 * kh;               // 0..4095
    int b_i = row >> 9, s = row & 511;
    dst[(((size_t)(b_i * NH + h) * SEQ + s) << 6) + d] = c[r] + bval;
  }
}

// ---------------------------------------------------------------------------
// Kernel 2: qrel[bh,s,r] = q[bh,s,:] . tab[r,:]   (M=49152, N=144, K=64)
// tab row-major == B^T, so both fragments are contiguous float2 loads.
// grid (768, 9), block 128 (4 waves, waves split M)
// ---------------------------------------------------------------------------
__global__ void k_qrel(const float* __restrict__ q, const float* __restrict__ tab,
                       float* __restrict__ qrel) {
  const int wave = threadIdx.x >> 5;
  const int lane = threadIdx.x & 31;
  const int lm = lane & 15, kh = lane >> 4;
  const int m0 = (blockIdx.x * 4 + wave) * 16;   // row in [0,49152)
  const int n0 = blockIdx.y * 16;

  const float* arow = q + (size_t)(m0 + lm) * HD + 2 * kh;
  const float* brow = tab + (size_t)(n0 + lm) * HD + 2 * kh;

  v8f c = ZERO8;
#pragma unroll
  for (int kk = 0; kk < HD; kk += 4) {
    v2f a = *(const v2f*)(arow + kk);
    v2f b = *(const v2f*)(brow + kk);
    c = wmma_f32(a, b, c);
  }
#pragma unroll
  for (int r = 0; r < 8; ++r)
    qrel[(size_t)(m0 + r + 8 * kh) * RPAD + n0 + lm] = c[r];
}

// ---------------------------------------------------------------------------
// Kernel 3: fused attention per (bh, 32-row query block).
// Wave-private 16-row tile: scores(WMMA K=64) + qrel gather -> softmax ->
// prel -> ctx = probs@v (K=512) + prel@tab (K=144), both WMMA, shared accums.
// grid (96, 16), block 64 (2 waves), dynamic LDS 84992 B
// ---------------------------------------------------------------------------
#define LDS_S 516   // 512 + 4  : lane-m A-frag reads land on distinct banks
#define LDS_P 148   // 144 + 4
#define NWAVE 2

__global__ void k_attn(const float* __restrict__ q, const float* __restrict__ k,
                       const float* __restrict__ v, const float* __restrict__ tab,
                       const float* __restrict__ qrel, float* __restrict__ ctx) {
  extern __shared__ float smem[];
  const int wave = threadIdx.x >> 5;
  const int lane = threadIdx.x & 31;
  const int lm = lane & 15, kh = lane >> 4;
  const int bh = blockIdx.x;                       // 0..95
  const int s0 = blockIdx.y * (16 * NWAVE) + wave * 16;

  float* sc = smem + (size_t)wave * 16 * LDS_S;                    // probs
  float* pr = smem + (size_t)NWAVE * 16 * LDS_S + (size_t)wave * 16 * LDS_P;

  // ---- Phase A: scores = q @ k^T, + qrel gather, * 1/sqrt(64) -> LDS ----
  const float* kbase = k + (size_t)bh * SEQ * HD + 2 * kh;
  v2f af[16];
  {
    const float* arow = q + (size_t)(bh * SEQ + s0 + lm) * HD + 2 * kh;
#pragma unroll
    for (int kk = 0; kk < 16; ++kk) af[kk] = *(const v2f*)(arow + 4 * kk);
  }
  for (int j = 0; j < 32; ++j) {
    const float* brow = kbase + (size_t)(j * 16 + lm) * HD;
    v8f c = ZERO8;
#pragma unroll
    for (int kk = 0; kk < 16; ++kk) {
      v2f b = *(const v2f*)(brow + 4 * kk);
      c = wmma_f32(af[kk], b, c);
    }
    const int t = j * 16 + lm;
#pragma unroll
    for (int r = 0; r < 8; ++r) {
      int srow = r + 8 * kh;
      int s = s0 + srow;
      int rr = t - s;
      rr = (rr < -64) ? -64 : (rr > 64 ? 64 : rr);
      rr += 64;
      sc[srow * LDS_S + t] =
          (c[r] + qrel[(size_t)(bh * SEQ + s) * RPAD + rr]) * 0.125f;
    }
  }
  __syncthreads();

  // ---- Phase B: softmax over 512 cols, wave-cooperative per row ----
  for (int row = 0; row < 16; ++row) {
    float* rp = sc + row * LDS_S;
    float mx = -3.4e38f;
    for (int t = lane; t < SEQ; t += 32) mx = fmaxf(mx, rp[t]);
#pragma unroll
    for (int off = 16; off > 0; off >>= 1) mx = fmaxf(mx, __shfl_xor(mx, off, 32));
    float sum = 0.f;
    for (int t = lane; t < SEQ; t += 32) {
      float e = __expf(rp[t] - mx);
      rp[t] = e;
      sum += e;
    }
#pragma unroll
    for (int off = 16; off > 0; off >>= 1) sum += __shfl_xor(sum, off, 32);
    float inv = 1.f / sum;
    for (int t = lane; t < SEQ; t += 32) rp[t] *= inv;
  }

  // ---- Phase C: prel[s,r] = sum of probs in each clipped-distance bucket ----
  for (int row = 0; row < 16; ++row) {
    const int s = s0 + row;
    const float* rp = sc + row * LDS_S;
    float* pp = pr + row * LDS_P;
    for (int r = lane; r < LDS_P; r += 32) {
      float val = 0.f;
      if (r >= 1 && r <= 127) {
        int t = s + r - 64;
        if (t >= 0 && t < SEQ) val = rp[t];
      }
      pp[r] = val;
    }
    float a0 = 0.f;                         // r=0 bucket: t <= s-64
    for (int t = lane; t <= s - 64; t += 32) a0 += rp[t];
#pragma unroll
    for (int off = 16; off > 0; off >>= 1) a0 += __shfl_xor(a0, off, 32);
    float a1 = 0.f;                         // r=128 bucket: t >= s+64
    for (int t = s + 64 + lane; t < SEQ; t += 32) a1 += rp[t];
#pragma unroll
    for (int off = 16; off > 0; off >>= 1) a1 += __shfl_xor(a1, off, 32);
    if (lane == 0) { pp[0] = a0; pp[128] = a1; }
  }
  __syncthreads();

  // ---- Phase D: ctx = probs @ v (K=512)  +  prel @ tab (K=144) ----
  v8f c0 = ZERO8, c1 = ZERO8, c2 = ZERO8, c3 = ZERO8;
  const float* vb = v + (size_t)bh * SEQ * HD;
  for (int kk = 0; kk < SEQ; kk += 4) {
    const int kb = kk + 2 * kh;
    v2f a = *(const v2f*)(sc + lm * LDS_S + kb);
    const float* r0 = vb + (size_t)kb * HD;
    const float* r1 = r0 + HD;
    v2f b;
    b.x = r0[lm];      b.y = r1[lm];      c0 = wmma_f32(a, b, c0);
    b.x = r0[lm + 16]; b.y = r1[lm + 16]; c1 = wmma_f32(a, b, c1);
    b.x = r0[lm + 32]; b.y = r1[lm + 32]; c2 = wmma_f32(a, b, c2);
    b.x = r0[lm + 48]; b.y = r1[lm + 48]; c3 = wmma_f32(a, b, c3);
  }
  for (int kk = 0; kk < RPAD; kk += 4) {
    const int kb = kk + 2 * kh;
    v2f a = *(const v2f*)(pr + lm * LDS_P + kb);
    const float* r0 = tab + (size_t)kb * HD;
    const float* r1 = r0 + HD;
    v2f b;
    b.x = r0[lm];      b.y = r1[lm];      c0 = wmma_f32(a, b, c0);
    b.x = r0[lm + 16]; b.y = r1[lm + 16]; c1 = wmma_f32(a, b, c1);
    b.x = r0[lm + 32]; b.y = r1[lm + 32]; c2 = wmma_f32(a, b, c2);
    b.x = r0[lm + 48]; b.y = r1[lm + 48]; c3 = wmma_f32(a, b, c3);
  }

  const int b_i = bh / NH, h = bh % NH;
  float* obase = ctx + (size_t)b_i * SEQ * HID + h * HD;
#pragma unroll
  for (int r = 0; r < 8; ++r) {
    int s = s0 + r + 8 * kh;
    float* o = obase + (size_t)s * HID;
    o[lm]      = c0[r];
    o[lm + 16] = c1[r];
    o[lm + 32] = c2[r];
    o[lm + 48] = c3[r];
  }
}

// ---------------------------------------------------------------------------
// Kernel 4: mean-pool over sequence: x[b,c] = mean_s ctx[b,s,c]
// ---------------------------------------------------------------------------
__global__ void k_pool(const float* __restrict__ ctx, float* __restrict__ x) {
  int idx = blockIdx.x * 256 + threadIdx.x;
  if (idx >= BATCH * HID) return;
  int b = idx / HID, cc = idx % HID;
  const float* p = ctx + (size_t)b * SEQ * HID + cc;
  float acc = 0.f;
  for (int s = 0; s < SEQ; ++s) acc += p[(size_t)s * HID];
  x[idx] = acc * (1.f / (float)SEQ);
}

// ---------------------------------------------------------------------------
// Kernel 5: MMoE + per-task DNNs.  B=8 -> far below matrix-op granularity;
// single workgroup, LDS-staged phases.
// ---------------------------------------------------------------------------
__global__ void k_mmoe(const float* __restrict__ x,
                       const float* __restrict__ experts,
                       const float* __restrict__ experts_bias,
                       const float* __restrict__ gates,
                       const float* __restrict__ gates_bias,
                       const float* __restrict__ W1, const float* __restrict__ b1,
                       const float* __restrict__ W2, const float* __restrict__ b2,
                       float* __restrict__ out) {
  __shared__ float eo[BATCH * MMOE_DIM * N_EXPERT];   // 12 KB
  __shared__ float gt[N_TASK * BATCH * N_EXPERT];     // tiny
  __shared__ float ti[N_TASK * BATCH * MMOE_DIM];     // 16 KB
  __shared__ float h1[N_TASK * BATCH * DNN_DIM];      // 16 KB
  const int tid = threadIdx.x;

  // experts_out[b,m,e]
  for (int idx = tid; idx < BATCH * MMOE_DIM * N_EXPERT; idx += 256) {
    int b = idx / (MMOE_DIM * N_EXPERT);
    int rem = idx % (MMOE_DIM * N_EXPERT);
    int m = rem / N_EXPERT, e = rem % N_EXPERT;
    float acc = experts_bias[m * N_EXPERT + e];
    const float* xb = x + b * HID;
    for (int hh = 0; hh < HID; ++hh)
      acc += xb[hh] * experts[(size_t)(hh * MMOE_DIM + m) * N_EXPERT + e];
    eo[idx] = acc;
  }
  // gate logits[t,b,e]
  for (int idx = tid; idx < N_TASK * BATCH * N_EXPERT; idx += 256) {
    int t = idx / (BATCH * N_EXPERT);
    int rem = idx % (BATCH * N_EXPERT);
    int b = rem / N_EXPERT, e = rem % N_EXPERT;
    float acc = gates_bias[t * N_EXPERT + e];
    const float* xb = x + b * HID;
    for (int hh = 0; hh < HID; ++hh)
      acc += xb[hh] * gates[(size_t)(t * HID + hh) * N_EXPERT + e];
    gt[idx] = acc;
  }
  __syncthreads();
  // softmax over experts
  if (tid < N_TASK * BATCH) {
    float* g = gt + tid * N_EXPERT;
    float m = fmaxf(g[0], fmaxf(g[1], g[2]));
    float e0 = __expf(g[0] - m), e1 = __expf(g[1] - m), e2 = __expf(g[2] - m);
    float inv = 1.f / (e0 + e1 + e2);
    g[0] = e0 * inv; g[1] = e1 * inv; g[2] = e2 * inv;
  }
  __syncthreads();
  // task_in[t,b,m]
  for (int idx = tid; idx < N_TASK * BATCH * MMOE_DIM; idx += 256) {
    int t = idx / (BATCH * MMOE_DIM);
    int b = (idx / MMOE_DIM) % BATCH;
    int m = idx % MMOE_DIM;
    const float* g = gt + (t * BATCH + b) * N_EXPERT;
    const float* e = eo + (b * MMOE_DIM + m) * N_EXPERT;
    ti[idx] = g[0] * e[0] + g[1] * e[1] + g[2] * e[2];
  }
  __syncthreads();
  // h1 = relu(task_in @ W1 + b1)
  for (int idx = tid; idx < N_TASK * BATCH * DNN_DIM; idx += 256) {
    int t = idx / (BATCH * DNN_DIM);
    int b = (idx / DNN_DIM) % BATCH;
    int d = idx % DNN_DIM;
    float acc = b1[t * DNN_DIM + d];
    const float* tb = ti + (t * BATCH + b) * MMOE_DIM;
    const float* w = W1 + (size_t)t * MMOE_DIM * DNN_DIM + d;
    for (int m = 0; m < MMOE_DIM; ++m) acc += tb[m] * w[(size_t)m * DNN_DIM];
    h1[idx] = fmaxf(acc, 0.f);
  }
  __syncthreads();
  // out = relu(h1 @ W2 + b2)
  for (int idx = tid; idx < N_TASK * BATCH * DNN_DIM; idx += 256) {
    int t = idx / (BATCH * DNN_DIM);
    int b = (idx / DNN_DIM) % BATCH;
    int d = idx % DNN_DIM;
    float acc = b2[t * DNN_DIM + d];
    const float* hb = h1 + (t * BATCH + b) * DNN_DIM;
    const float* w = W2 + (size_t)t * DNN_DIM * DNN_DIM + d;
    for (int m = 0; m < DNN_DIM; ++m) acc += hb[m] * w[(size_t)m * DNN_DIM];
    out[idx] = fmaxf(acc, 0.f);
  }
}

// ---------------------------------------------------------------------------
extern "C" void kernel_launch(void* const* d_in, const int* in_sizes, int n_in,
                              void* d_out, int out_size, void* d_ws,
                              size_t ws_size, hipStream_t stream) {
  (void)in_sizes; (void)n_in; (void)out_size; (void)ws_size;

  const float* hs  = (const float*)d_in[0];
  const float* Wq  = (const float*)d_in[1];
  const float* bq  = (const float*)d_in[2];
  const float* Wk  = (const float*)d_in[3];
  const float* bk  = (const float*)d_in[4];
  const float* Wv  = (const float*)d_in[5];
  const float* bv  = (const float*)d_in[6];
  const float* experts      = (const float*)d_in[7];
  const float* experts_bias = (const float*)d_in[8];
  const float* gates        = (const float*)d_in[9];
  const float* gates_bias   = (const float*)d_in[10];
  const float* dW1 = (const float*)d_in[11];
  const float* db1 = (const float*)d_in[12];
  const float* dW2 = (const float*)d_in[13];
  const float* db2 = (const float*)d_in[14];
  float* out = (float*)d_out;

  float* ws   = (float*)d_ws;
  float* tab  = ws + OFF_TAB;
  float* q    = ws + OFF_Q;
  float* k    = ws + OFF_K;
  float* v    = ws + OFF_V;
  float* qrel = ws + OFF_QREL;
  float* ctx  = ws + OFF_CTX;
  float* x    = ws + OFF_X;

  // 0) rel-pos table
  k_tab<<<dim3((RPAD * HD + 255) / 256), dim3(256), 0, stream>>>(tab);

  // 1) QKV projections -> head-major [BH,S,HD]
  k_qkv<<<dim3(256, 12, 3), dim3(128), 0, stream>>>(hs, Wq, bq, Wk, bk, Wv, bv,
                                                    q, k, v);

  // 2) qrel = q @ tab^T
  k_qrel<<<dim3(768, 9), dim3(128), 0, stream>>>(q, tab, qrel);

  // 3) fused attention (2 waves/WG, 84992 B dynamic LDS)
  const size_t smem =
      (size_t)(NWAVE * 16 * LDS_S + NWAVE * 16 * LDS_P) * sizeof(float);
  k_attn<<<dim3(BH, SEQ / (16 * NWAVE)), dim3(32 * NWAVE), smem, stream>>>(
      q, k, v, tab, qrel, ctx);

  // 4) mean pool
  k_pool<<<dim3((BATCH * HID + 255) / 256), dim3(256), 0, stream>>>(ctx, x);

  // 5) MMoE + task DNNs
  k_mmoe<<<dim3(1), dim3(256), 0, stream>>>(x, experts, experts_bias, gates,
                                            gates_bias, dW1, db1, dW2, db2,
                                            out);
}